// GCN2Conv_47648367182322
// MI455X (gfx1250) — compile-verified
//
#include <hip/hip_runtime.h>
#include <hip/hip_bf16.h>
#include <math.h>

#define D 64
#define ALPHA 0.1f
#define BETA  0.6931471805599453f   // log(2)

typedef float v2f __attribute__((ext_vector_type(2)));
typedef float v8f __attribute__((ext_vector_type(8)));

// ---------------- degree count ----------------
__global__ void gcn2_degree_kernel(const int* __restrict__ dst,
                                   unsigned* __restrict__ deg, int e) {
    int i = blockIdx.x * blockDim.x + threadIdx.x;
    if (i < e) atomicAdd(&deg[dst[i]], 1u);
}

// ---------------- norm = rsqrt(max(deg,1)) ----------------
__global__ void gcn2_norm_kernel(const unsigned* __restrict__ deg,
                                 float* __restrict__ norm, int n) {
    int i = blockIdx.x * blockDim.x + threadIdx.x;
    if (i < n) {
        float d = (float)deg[i];
        norm[i] = rsqrtf(fmaxf(d, 1.0f));
    }
}

// ---------------- h = feat * norm[row]  (float4 vectorized) ----------------
__global__ void gcn2_scale_kernel(const float4* __restrict__ feat4,
                                  const float* __restrict__ norm,
                                  float4* __restrict__ h4, int n4) {
    int i = blockIdx.x * blockDim.x + threadIdx.x;   // n4 = n * 16
    if (i < n4) {
        float s = norm[i >> 4];
        float4 v = feat4[i];
        v.x *= s; v.y *= s; v.z *= s; v.w *= s;
        h4[i] = v;
    }
}

// ---------------- SpMM scatter: agg[dst] += h[src]  (16 lanes/edge, float4) --
__global__ void gcn2_scatter_kernel(const int* __restrict__ src,
                                    const int* __restrict__ dst,
                                    const float4* __restrict__ h4,
                                    float* __restrict__ agg, int total) {
    int i = blockIdx.x * blockDim.x + threadIdx.x;   // total = e * 16
    if (i < total) {
        int e = i >> 4;
        int p = i & 15;
        int s = src[e];
        int d = dst[e];
        float4 v = h4[(size_t)s * 16 + p];
        float* base = agg + (size_t)d * D + p * 4;
        __hip_atomic_fetch_add(base + 0, v.x, __ATOMIC_RELAXED, __HIP_MEMORY_SCOPE_AGENT);
        __hip_atomic_fetch_add(base + 1, v.y, __ATOMIC_RELAXED, __HIP_MEMORY_SCOPE_AGENT);
        __hip_atomic_fetch_add(base + 2, v.z, __ATOMIC_RELAXED, __HIP_MEMORY_SCOPE_AGENT);
        __hip_atomic_fetch_add(base + 3, v.w, __ATOMIC_RELAXED, __HIP_MEMORY_SCOPE_AGENT);
    }
}

// ---------------- fused finalize + projection via WMMA f32 16x16x4 ----------
// One wave32 per 16-row tile. feat_sum tile staged in LDS, then
//   D = feat_sum @ (BETA*W)  +  [ (1-BETA)*feat_sum + bias ]   == rst
__launch_bounds__(32)
__global__ void gcn2_project_kernel(const float* __restrict__ agg,
                                    const float* __restrict__ feat0,
                                    const float* __restrict__ norm,
                                    const float* __restrict__ W,
                                    const float* __restrict__ bias,
                                    float* __restrict__ out, int n) {
    __shared__ float fs[16 * D];          // 4 KB feat_sum tile
    const int t = threadIdx.x;            // 0..31
    const int row_base = blockIdx.x * 16;

    // Stage feat_sum = 0.9*agg*norm + 0.1*feat_0 into LDS (coalesced float4)
    #pragma unroll
    for (int i = 0; i < 8; ++i) {
        int q  = t + 32 * i;              // float4 slot 0..255 within tile
        int r  = q >> 4;                  // tile row 0..15
        int c4 = q & 15;                  // float4 column
        int g  = row_base + r;
        if (g > n - 1) g = n - 1;         // clamp (stores are guarded later)
        float4 a4 = ((const float4*)agg)[(size_t)g * 16 + c4];
        float4 f4 = ((const float4*)feat0)[(size_t)g * 16 + c4];
        float  s  = norm[g] * (1.0f - ALPHA);
        float4 v;
        v.x = fmaf(a4.x, s, ALPHA * f4.x);
        v.y = fmaf(a4.y, s, ALPHA * f4.y);
        v.z = fmaf(a4.z, s, ALPHA * f4.z);
        v.w = fmaf(a4.w, s, ALPHA * f4.w);
        ((float4*)fs)[q] = v;
    }
    __syncthreads();

    const int h = t >> 4;                 // half-wave select (K phase)
    const int m = t & 15;                 // row (A) / column (B,C,D) index

    // A fragments: lane holds A[m][4*ki+2h], A[m][4*ki+2h+1]  (ISA 16x4 f32 layout)
    v2f a[16];
    #pragma unroll
    for (int ki = 0; ki < 16; ++ki)
        a[ki] = *(const v2f*)&fs[m * D + 4 * ki + 2 * h];

    #pragma unroll
    for (int nt = 0; nt < 4; ++nt) {
        const int n0  = nt * 16;
        const int col = n0 + m;
        const float bcol = bias[col];

        // C seed = (1-BETA)*feat_sum + bias in C/D layout (VGPR r: M = 8h + r)
        v8f c;
        #pragma unroll
        for (int r = 0; r < 8; ++r)
            c[r] = fmaf(1.0f - BETA, fs[(8 * h + r) * D + col], bcol);

        // Accumulate feat_sum @ (BETA*W) over K in steps of 4
        #pragma unroll
        for (int ki = 0; ki < 16; ++ki) {
            int k0 = 4 * ki + 2 * h;
            v2f b;
            b.x = BETA * W[(size_t)k0 * D + col];
            b.y = BETA * W[(size_t)(k0 + 1) * D + col];
            c = __builtin_amdgcn_wmma_f32_16x16x4_f32(
                    false, a[ki], false, b, (short)0, c, false, false);
        }

        // Store D tile (row-guarded stores only; WMMA above is uniform)
        #pragma unroll
        for (int r = 0; r < 8; ++r) {
            int grow = row_base + 8 * h + r;
            if (grow < n) out[(size_t)grow * D + col] = c[r];
        }
    }
}

extern "C" void kernel_launch(void* const* d_in, const int* in_sizes, int n_in,
                              void* d_out, int out_size, void* d_ws, size_t ws_size,
                              hipStream_t stream) {
    const float* feat  = (const float*)d_in[0];
    const float* feat0 = (const float*)d_in[1];
    const int*   src   = (const int*)  d_in[2];
    const int*   dst   = (const int*)  d_in[3];
    const float* W     = (const float*)d_in[4];
    const float* bias  = (const float*)d_in[5];
    float* out = (float*)d_out;

    const int n = in_sizes[0] / D;        // 100000
    const int e = in_sizes[2];            // 1280000

    // Workspace layout: [deg u32 x n][agg f32 x n*D][norm f32 x n][h f32 x n*D]
    char* ws = (char*)d_ws;
    unsigned* deg  = (unsigned*)ws;
    float*    agg  = (float*)(ws + sizeof(unsigned) * (size_t)n);
    float*    norm = (float*)(ws + sizeof(unsigned) * (size_t)n
                                 + sizeof(float) * (size_t)n * D);
    float*    h    = norm + n;

    // Zero deg + agg in one contiguous memset (graph-capturable)
    size_t zero_bytes = sizeof(unsigned) * (size_t)n + sizeof(float) * (size_t)n * D;
    hipMemsetAsync(ws, 0, zero_bytes, stream);

    gcn2_degree_kernel<<<(e + 255) / 256, 256, 0, stream>>>(dst, deg, e);
    gcn2_norm_kernel<<<(n + 255) / 256, 256, 0, stream>>>(deg, norm, n);

    int n4 = n * 16;
    gcn2_scale_kernel<<<(n4 + 255) / 256, 256, 0, stream>>>(
        (const float4*)feat, norm, (float4*)h, n4);

    int total = e * 16;
    gcn2_scatter_kernel<<<(total + 255) / 256, 256, 0, stream>>>(
        src, dst, (const float4*)h, agg, total);

    int tiles = (n + 15) / 16;
    gcn2_project_kernel<<<tiles, 32, 0, stream>>>(agg, feat0, norm, W, bias, out, n);
}